// FCLTCQNetwork_23106924052951
// MI455X (gfx1250) — compile-verified
//
#include <hip/hip_runtime.h>
#include <hip/hip_bf16.h>

// Problem sizes (fixed by the reference)
#define B_  4096
#define T_  128
#define D_  64
#define U_  64
#define A_  8
#define QH_ 64
#define UNF 6
#define EPS_ 1e-8f

#define ROWS 8                  // batch rows per block
#define THREADS (ROWS * 64)     // one thread per (row, neuron)

typedef __attribute__((ext_vector_type(2))) float v2f;
typedef __attribute__((ext_vector_type(8))) float v8f;

__device__ __forceinline__ float softplus_f(float x) {
    return (x > 20.0f) ? x : log1pf(__expf(x));
}

// tanh via the gfx1250 hardware trans op when available (1 trans op),
// otherwise exp2+rcp (2 trans ops).
__device__ __forceinline__ float fast_tanh(float x) {
#if __has_builtin(__builtin_amdgcn_tanhf)
    return __builtin_amdgcn_tanhf(x);
#else
    // tanh(x) = 1 - 2/(exp(2x)+1)
    float t = __builtin_amdgcn_exp2f(x * 2.88539008177792681472f); // exp(2x)
    return 1.0f - 2.0f * __builtin_amdgcn_rcpf(t + 1.0f);
#endif
}

// ---------------------------------------------------------------------------
// LTC recurrence: grid = B/ROWS blocks, 512 threads. Thread (r, j) owns the
// hidden state v[b0+r][j].
//
// Using sigmoid(z) = 1/2 + 1/2*tanh(z/2), each synapse contribution
//   wp*sigmoid(sigma*(v_i - mu)) = wp/2 + (wp/2)*tanh( (sigma/2)*v_i - sigma*mu/2 )
// so we pack params as (sigma*mu/2, sigma/2, wp/2, wp*erev/2) and hoist the
// constant sum of wp/2 (resp. wp*erev/2) columns out of the loop entirely.
// Inner loop per synapse: 1 ds_load_b128 (amortized +1/4 ds_load_b128 for the
// v broadcast) + 1 fma + 1 v_tanh_f32 + 2 fma.
// ---------------------------------------------------------------------------
__global__ __launch_bounds__(THREADS)
void ltc_recurrence_kernel(const float* __restrict__ obs,
                           const float* __restrict__ input_w,
                           const float* __restrict__ input_b,
                           const float* __restrict__ s_mu,
                           const float* __restrict__ s_sigma,
                           const float* __restrict__ s_w,
                           const float* __restrict__ s_erev,
                           const float* __restrict__ r_mu,
                           const float* __restrict__ r_sigma,
                           const float* __restrict__ r_w,
                           const float* __restrict__ r_erev,
                           const float* __restrict__ gleak,
                           const float* __restrict__ vleak,
                           const float* __restrict__ cm,
                           float* __restrict__ hidden_out)
{
    extern __shared__ float4 smem4[];
    float4* rec = smem4;                      // [U_*U_] recurrent params
    float4* sen = smem4 + U_ * U_;            // [D_*U_] sensory params
    float*  xsh = (float*)(smem4 + U_*U_ + D_*U_);  // [ROWS*64] staged obs (16B aligned)
    float*  vsh = xsh + ROWS * 64;                  // [ROWS*64] broadcast v (16B aligned)

    const int tid = threadIdx.x;
    const int r   = tid >> 6;       // 0..ROWS-1
    const int j   = tid & 63;       // neuron / input-channel index
    const int b   = blockIdx.x * ROWS + r;

    // Pack parameter tables into LDS (softplus + half-scales folded in once).
    for (int idx = tid; idx < U_ * U_; idx += THREADS) {
        float mu = r_mu[idx], sg = r_sigma[idx];
        float wph = 0.5f * softplus_f(r_w[idx]);
        rec[idx] = make_float4(0.5f * sg * mu, 0.5f * sg, wph, wph * r_erev[idx]);
    }
    for (int idx = tid; idx < D_ * U_; idx += THREADS) {
        float mu = s_mu[idx], sg = s_sigma[idx];
        float wph = 0.5f * softplus_f(s_w[idx]);
        sen[idx] = make_float4(0.5f * sg * mu, 0.5f * sg, wph, wph * s_erev[idx]);
    }
    const float iw  = input_w[j];
    const float ib  = input_b[j];
    const float gl  = softplus_f(gleak[j]);
    const float glv = gl * vleak[j];
    const float cmt = softplus_f(cm[j]) * (float)UNF;
    __syncthreads();

    // Hoisted constant halves: sum over presynaptic index of wp/2 and wp*erev/2.
    float rec_cd = 0.0f, rec_cn = 0.0f, sen_cd = 0.0f, sen_cn = 0.0f;
    for (int i = 0; i < U_; ++i) { float4 p = rec[i * 64 + j]; rec_cd += p.z; rec_cn += p.w; }
    for (int d = 0; d < D_; ++d) { float4 p = sen[d * 64 + j]; sen_cd += p.z; sen_cn += p.w; }

    const float* obs_b = obs + (size_t)b * T_ * D_;

    float v = 0.0f;

    for (int t = 0; t < T_; ++t) {
        // Stage this timestep's observation (input affine applied; d == j here).
        xsh[r * 64 + j] = fmaf(obs_b[t * D_ + j], iw, ib);
        if (t + 1 < T_) __builtin_prefetch(&obs_b[(t + 1) * D_ + j], 0, 0);
        __syncthreads();

        // Sensory synapses: reduce over D into (w_num_s, w_den_s) for (r, j).
        float wn = sen_cn, wd = sen_cd;
        #pragma unroll 4
        for (int d4 = 0; d4 < D_ / 4; ++d4) {
            const float4 xv = *(const float4*)&xsh[r * 64 + d4 * 4];
            float4 p0 = sen[(d4 * 4 + 0) * 64 + j];
            float4 p1 = sen[(d4 * 4 + 1) * 64 + j];
            float4 p2 = sen[(d4 * 4 + 2) * 64 + j];
            float4 p3 = sen[(d4 * 4 + 3) * 64 + j];
            float t0 = fast_tanh(fmaf(p0.y, xv.x, -p0.x));
            float t1 = fast_tanh(fmaf(p1.y, xv.y, -p1.x));
            float t2 = fast_tanh(fmaf(p2.y, xv.z, -p2.x));
            float t3 = fast_tanh(fmaf(p3.y, xv.w, -p3.x));
            wd = fmaf(p0.z, t0, wd); wn = fmaf(p0.w, t0, wn);
            wd = fmaf(p1.z, t1, wd); wn = fmaf(p1.w, t1, wn);
            wd = fmaf(p2.z, t2, wd); wn = fmaf(p2.w, t2, wn);
            wd = fmaf(p3.z, t3, wd); wn = fmaf(p3.w, t3, wn);
        }

        // ODE unfolds.
        for (int u = 0; u < UNF; ++u) {
            vsh[r * 64 + j] = v;
            __syncthreads();
            float num = fmaf(cmt, v, glv) + wn + rec_cn;
            float den = cmt + gl + wd + rec_cd;
            #pragma unroll 4
            for (int i4 = 0; i4 < U_ / 4; ++i4) {
                const float4 vv = *(const float4*)&vsh[r * 64 + i4 * 4];
                float4 p0 = rec[(i4 * 4 + 0) * 64 + j];
                float4 p1 = rec[(i4 * 4 + 1) * 64 + j];
                float4 p2 = rec[(i4 * 4 + 2) * 64 + j];
                float4 p3 = rec[(i4 * 4 + 3) * 64 + j];
                float t0 = fast_tanh(fmaf(p0.y, vv.x, -p0.x));
                float t1 = fast_tanh(fmaf(p1.y, vv.y, -p1.x));
                float t2 = fast_tanh(fmaf(p2.y, vv.z, -p2.x));
                float t3 = fast_tanh(fmaf(p3.y, vv.w, -p3.x));
                den = fmaf(p0.z, t0, den); num = fmaf(p0.w, t0, num);
                den = fmaf(p1.z, t1, den); num = fmaf(p1.w, t1, num);
                den = fmaf(p2.z, t2, den); num = fmaf(p2.w, t2, num);
                den = fmaf(p3.z, t3, den); num = fmaf(p3.w, t3, num);
            }
            v = num / (den + EPS_);    // precise division: only 6/timestep/thread
            __syncthreads();
        }
    }
    hidden_out[(size_t)b * U_ + j] = v;
}

// ---------------------------------------------------------------------------
// MLP head with fp32 WMMA (V_WMMA_F32_16X16X4_F32, K stepped by 4).
// One wave per 16 batch rows. A-layout: lane l -> M=l%16, K pair (l>=16)*2.
// B/C/D layout: lane l -> N=l%16, rows split across lane halves / VGPRs.
// ---------------------------------------------------------------------------
__global__ __launch_bounds__(32)
void ltc_head_kernel(const float* __restrict__ hidden,
                     const float* __restrict__ W1,
                     const float* __restrict__ b1,
                     const float* __restrict__ W2,
                     const float* __restrict__ b2,
                     float* __restrict__ q)
{
    __shared__ float h1sh[16 * QH_];

    const int lane = threadIdx.x;
    const int m    = lane & 15;
    const int hi   = lane >> 4;          // which K/M half this lane serves
    const int row0 = blockIdx.x * 16;

    // ---- GEMM 1: hidden(16x64) @ W1(64x64), 4 N-tiles of 16 ----
    v8f acc0 = {}, acc1 = {}, acc2_ = {}, acc3 = {};
    const float* Arow = hidden + (size_t)(row0 + m) * U_;
    for (int k0 = 0; k0 < U_; k0 += 4) {
        const int ka = k0 + 2 * hi;
        v2f a;  a.x = Arow[ka];  a.y = Arow[ka + 1];
        v2f b0v; b0v.x = W1[ka * QH_ +  0 + m]; b0v.y = W1[(ka + 1) * QH_ +  0 + m];
        v2f b1v; b1v.x = W1[ka * QH_ + 16 + m]; b1v.y = W1[(ka + 1) * QH_ + 16 + m];
        v2f b2v; b2v.x = W1[ka * QH_ + 32 + m]; b2v.y = W1[(ka + 1) * QH_ + 32 + m];
        v2f b3v; b3v.x = W1[ka * QH_ + 48 + m]; b3v.y = W1[(ka + 1) * QH_ + 48 + m];
        acc0  = __builtin_amdgcn_wmma_f32_16x16x4_f32(false, a, false, b0v, (short)0, acc0,  false, false);
        acc1  = __builtin_amdgcn_wmma_f32_16x16x4_f32(false, a, false, b1v, (short)0, acc1,  false, false);
        acc2_ = __builtin_amdgcn_wmma_f32_16x16x4_f32(false, a, false, b2v, (short)0, acc2_, false, false);
        acc3  = __builtin_amdgcn_wmma_f32_16x16x4_f32(false, a, false, b3v, (short)0, acc3,  false, false);
    }

    // bias + relu, scatter C-layout -> row-major LDS tile
    #pragma unroll
    for (int g = 0; g < 8; ++g) {
        const int row = hi * 8 + g;
        { float h = acc0[g]  + b1[ 0 + m]; h1sh[row * QH_ +  0 + m] = fmaxf(h, 0.0f); }
        { float h = acc1[g]  + b1[16 + m]; h1sh[row * QH_ + 16 + m] = fmaxf(h, 0.0f); }
        { float h = acc2_[g] + b1[32 + m]; h1sh[row * QH_ + 32 + m] = fmaxf(h, 0.0f); }
        { float h = acc3[g]  + b1[48 + m]; h1sh[row * QH_ + 48 + m] = fmaxf(h, 0.0f); }
    }
    __syncthreads();

    // ---- GEMM 2: h1(16x64) @ W2(64x8), N padded to 16 with zero columns ----
    v8f acc = {};
    for (int k0 = 0; k0 < QH_; k0 += 4) {
        const int ka = k0 + 2 * hi;
        v2f a;  a.x = h1sh[m * QH_ + ka];  a.y = h1sh[m * QH_ + ka + 1];
        v2f bb;
        bb.x = (m < A_) ? W2[ka * A_ + m]       : 0.0f;
        bb.y = (m < A_) ? W2[(ka + 1) * A_ + m] : 0.0f;
        acc = __builtin_amdgcn_wmma_f32_16x16x4_f32(false, a, false, bb, (short)0, acc, false, false);
    }
    if (m < A_) {
        #pragma unroll
        for (int g = 0; g < 8; ++g) {
            const int row = row0 + hi * 8 + g;
            q[(size_t)row * A_ + m] = acc[g] + b2[m];
        }
    }
}

// ---------------------------------------------------------------------------
extern "C" void kernel_launch(void* const* d_in, const int* in_sizes, int n_in,
                              void* d_out, int out_size, void* d_ws, size_t ws_size,
                              hipStream_t stream) {
    const float* obs      = (const float*)d_in[0];
    const float* input_w  = (const float*)d_in[1];
    const float* input_b  = (const float*)d_in[2];
    const float* s_mu     = (const float*)d_in[3];
    const float* s_sigma  = (const float*)d_in[4];
    const float* s_w      = (const float*)d_in[5];
    const float* s_erev   = (const float*)d_in[6];
    const float* r_mu     = (const float*)d_in[7];
    const float* r_sigma  = (const float*)d_in[8];
    const float* r_w      = (const float*)d_in[9];
    const float* r_erev   = (const float*)d_in[10];
    const float* gleak    = (const float*)d_in[11];
    const float* vleak    = (const float*)d_in[12];
    const float* cm       = (const float*)d_in[13];
    const float* W1       = (const float*)d_in[14];
    const float* b1       = (const float*)d_in[15];
    const float* W2       = (const float*)d_in[16];
    const float* b2       = (const float*)d_in[17];

    float* q_out      = (float*)d_out;              // (B, A)
    float* hidden_out = q_out + (size_t)B_ * A_;    // (B, U), second tuple element

    const size_t lds_bytes = (size_t)(U_ * U_ + D_ * U_) * sizeof(float4)
                           + (size_t)2 * ROWS * 64 * sizeof(float);   // ~132 KB (<320 KB/WGP)

    ltc_recurrence_kernel<<<B_ / ROWS, THREADS, lds_bytes, stream>>>(
        obs, input_w, input_b, s_mu, s_sigma, s_w, s_erev,
        r_mu, r_sigma, r_w, r_erev, gleak, vleak, cm, hidden_out);

    ltc_head_kernel<<<B_ / 16, 32, 0, stream>>>(hidden_out, W1, b1, W2, b2, q_out);
}